// ShiftingLayer_14370960572815
// MI455X (gfx1250) — compile-verified
//
#include <hip/hip_runtime.h>
#include <stdint.h>

// Problem constants (from the reference): 4000x4000 fp32, CS=128, PAD=10.
#define PADR  10
#define CS    128
#define CSP   148            // CS + 2*PAD
#define LDSW  148            // LDS row stride in floats (592 B, 16B-aligned rows)
#define HW    4000
#define NT    32             // ceil(4000/128) tiles per dimension
#define ACOEF (-0.75f)

typedef __attribute__((ext_vector_type(4))) float        v4f;
typedef __attribute__((ext_vector_type(4))) unsigned int v4u;
typedef __attribute__((ext_vector_type(8))) int          v8i;
typedef __attribute__((ext_vector_type(4))) int          v4i;

__device__ __forceinline__ void cubic_w(float t, float& w0, float& w1, float& w2, float& w3) {
    float t2 = t * t, t3 = t2 * t;
    w0 = ACOEF * (t3 - 2.0f * t2 + t);
    w1 = (ACOEF + 2.0f) * t3 - (ACOEF + 3.0f) * t2 + 1.0f;
    float s = 1.0f - t;
    float s2 = s * s, s3 = s2 * s;
    w2 = (ACOEF + 2.0f) * s3 - (ACOEF + 3.0f) * s2 + 1.0f;
    w3 = ACOEF * (s3 - 2.0f * s2 + s);
}

__global__ __launch_bounds__(256) void bicubic_tile_kernel(
    const float* __restrict__ matrix,
    const float* __restrict__ theta,
    float* __restrict__ out)
{
    __shared__ __align__(16) float tile[CSP * LDSW];   // 87,616 B of the 320 KB WGP LDS

    const int tid = threadIdx.x;
    const int n   = blockIdx.x;          // tile id = i0*NT + i1
    const int i0  = n >> 5;
    const int i1  = n & (NT - 1);

    // Valid (in-image) sub-window of this tile's 148x148 halo window.
    // Source row of tile-row r is  i0*CS - 2*PAD + r  (likewise for columns).
    const int r0 = (i0 == 0) ? 2 * PADR : 0;
    const int r1 = min(CSP, HW + 2 * PADR - i0 * CS);  // 52 for i0==31, else 148
    const int c0 = (i1 == 0) ? 2 * PADR : 0;
    const int c1 = min(CSP, HW + 2 * PADR - i1 * CS);

    const bool border = (r0 != 0) || (r1 != CSP) || (c0 != 0) || (c1 != CSP);

    const long grb = (long)i0 * CS - 2 * PADR;  // global row of tile-row 0
    const long gcb = (long)i1 * CS - 2 * PADR;  // global col of tile-col 0

#if __has_builtin(__builtin_amdgcn_tensor_load_to_lds)
    if (!border) {
        // ---- Interior tile: one Tensor-Data-Mover DMA for the whole window ----
        // 2D D#: 148x148 fp32 tile, row stride 4000 floats, packed into LDS
        // (TDM packs rows contiguously == our LDSW=148 layout).
        if (tid < 32) {  // single wave issues the tensor op (EXEC ignored by TDM)
            uint64_t gaddr = (uint64_t)(uintptr_t)(matrix + grb * (long)HW + gcb);
            uint32_t laddr = (uint32_t)(uintptr_t)&tile[0];
            v4u g0 = { 1u,                                   // count=1, user mode
                       laddr,                                // lds_addr (bytes)
                       (uint32_t)gaddr,                      // global_addr[31:0]
                       ((uint32_t)(gaddr >> 32) & 0x01FFFFFFu) | 0x80000000u }; // [56:32] | type=2
            v8i g1 = { (int)0x00020000,      // data_size=2 (4B); wg_mask=0; no pad/iter
                       (int)(CSP << 16),     // tensor_dim0[15:0] << 16 (abar_addr=0)
                       (int)(CSP << 16),     // tensor_dim0[31:16]=0 | tensor_dim1[15:0]<<16
                       (int)(CSP << 16),     // tensor_dim1[31:16]=0 | tile_dim0<<16
                       (int)CSP,             // tile_dim1 | tile_dim2=0
                       (int)HW,              // tensor_dim0_stride[31:0]
                       0,                    // stride[47:32]=0 | dim1_stride lo=0
                       0 };                  // dim1_stride hi (unused for 2D)
            v4i g2 = { 0, 0, 0, 0 };
            v4i g3 = { 0, 0, 0, 0 };
            v8i g4 = { 0, 0, 0, 0, 0, 0, 0, 0 };
            __builtin_amdgcn_tensor_load_to_lds(g0, g1, g2, g3, g4, 0);
            __builtin_amdgcn_s_wait_tensorcnt(0);
        }
    } else
#endif
    {
        // ---- Border tile: zero-prefill, then per-lane async b128 copies ----
        v4f  z  = {0.0f, 0.0f, 0.0f, 0.0f};
        v4f* t4 = (v4f*)tile;
        for (int j = tid; j < (CSP * LDSW) / 4; j += 256) t4[j] = z;
        __syncthreads();  // zeros visible before async-engine writes land

        // All valid column ranges are multiples of 4 floats (20/52/128/148),
        // so b128 stays aligned on both the global and LDS side.
        const int nc4   = (c1 - c0) >> 2;
        const int total = (r1 - r0) * nc4;
        for (int q = tid; q < total; q += 256) {
            int r  = q / nc4;
            int cc = c0 + ((q - r * nc4) << 2);
            int rr = r0 + r;
            const float* gsrc = matrix + (grb + rr) * (long)HW + (gcb + cc);
            uint32_t lds = (uint32_t)(uintptr_t)&tile[rr * LDSW + cc];
            asm volatile("global_load_async_to_lds_b128 %0, %1, off"
                         :: "v"(lds), "v"(gsrc) : "memory");
        }
        asm volatile("s_wait_asynccnt 0x0" ::: "memory");
    }
    __syncthreads();

    // ---- Per-tile affine parameters (block-uniform) ----
    const float* th = theta + n * 6;
    const float t00 = th[0], t01 = th[1], t02 = th[2];
    const float t10 = th[3], t11 = th[4], t12 = th[5];

    const float inv_csp = 1.0f / (float)CSP;

    // ---- 64 output pixels per thread, 16-tap bicubic from LDS ----
    #pragma unroll 2
    for (int k = 0; k < (CS * CS) / 256; ++k) {
        int p  = tid + k * 256;
        int py = p >> 7;
        int px = p & (CS - 1);

        float xs = (2.0f * (float)(px + PADR) + 1.0f) * inv_csp - 1.0f;
        float ys = (2.0f * (float)(py + PADR) + 1.0f) * inv_csp - 1.0f;
        float gx = fmaf(t00, xs, fmaf(t01, ys, t02));
        float gy = fmaf(t10, xs, fmaf(t11, ys, t12));
        float ix = ((gx + 1.0f) * (float)CSP - 1.0f) * 0.5f;
        float iy = ((gy + 1.0f) * (float)CSP - 1.0f) * 0.5f;

        float fx = floorf(ix), fy = floorf(iy);
        float tx = ix - fx,    ty = iy - fy;
        int   x0 = (int)fx,    y0 = (int)fy;

        float wx[4], wy[4];
        cubic_w(tx, wx[0], wx[1], wx[2], wx[3]);
        cubic_w(ty, wy[0], wy[1], wy[2], wy[3]);

        // Fold tap validity into the weights (0-weight == masked tap),
        // clamp addresses once per column / row.
        int ccb[4];
        #pragma unroll
        for (int kx = 0; kx < 4; ++kx) {
            int col = x0 - 1 + kx;
            wx[kx]  = ((unsigned)col < (unsigned)CSP) ? wx[kx] : 0.0f;
            ccb[kx] = min(max(col, 0), CSP - 1);
        }

        float acc = 0.0f;
        #pragma unroll
        for (int ky = 0; ky < 4; ++ky) {
            int   row = y0 - 1 + ky;
            float wyk = ((unsigned)row < (unsigned)CSP) ? wy[ky] : 0.0f;
            int   rc  = min(max(row, 0), CSP - 1);
            const float* rp = &tile[rc * LDSW];
            float rs = fmaf(wx[0], rp[ccb[0]],
                       fmaf(wx[1], rp[ccb[1]],
                       fmaf(wx[2], rp[ccb[2]],
                            wx[3] * rp[ccb[3]])));
            acc = fmaf(wyk, rs, acc);
        }

        int gr = i0 * CS + py;
        int gc = i1 * CS + px;
        if (gr < HW && gc < HW)
            out[(long)gr * HW + gc] = acc;
    }
}

extern "C" void kernel_launch(void* const* d_in, const int* in_sizes, int n_in,
                              void* d_out, int out_size, void* d_ws, size_t ws_size,
                              hipStream_t stream) {
    (void)in_sizes; (void)n_in; (void)out_size; (void)d_ws; (void)ws_size;
    const float* matrix = (const float*)d_in[0];   // (4000, 4000) fp32
    const float* theta  = (const float*)d_in[1];   // (1024, 2, 3) fp32
    float*       out    = (float*)d_out;           // (4000, 4000) fp32

    dim3 grid(NT * NT);
    dim3 block(256);
    hipLaunchKernelGGL(bicubic_tile_kernel, grid, block, 0, stream, matrix, theta, out);
}